// GCNSegmentation_19705309954162
// MI455X (gfx1250) — compile-verified
//
#include <hip/hip_runtime.h>
#include <hip/hip_bf16.h>

#define N_NODES 10000
#define N_EDGES 320000
#define NODE_FEAT 2048
#define HID 256
#define LAYERS 3
#define NUM_CLASSES 10
#define BN_EPS 1e-5f
#define LDS_PAD 36   // dword stride: mult of 4 (b128-aligned), conflict-free columns

typedef float v2f __attribute__((ext_vector_type(2)));
typedef float v8f __attribute__((ext_vector_type(8)));
typedef int   v4i __attribute__((ext_vector_type(4)));
typedef __attribute__((address_space(3))) v4i lds_v4i;

#if __has_builtin(__builtin_amdgcn_global_load_async_to_lds_b128)
#define HAVE_ASYNC_LDS 1
#else
#define HAVE_ASYNC_LDS 0
#endif

#if __has_builtin(__builtin_amdgcn_s_wait_asynccnt)
#define WAIT_ASYNC(n) __builtin_amdgcn_s_wait_asynccnt(n)
#else
#define WAIT_ASYNC(n) asm volatile("s_wait_asynccnt %0" ::"i"(n) : "memory")
#endif

// ---------------------------------------------------------------------------
// Shared tile-compute: 32 x V_WMMA_F32_16X16X4_F32 over one 64x64x32 chunk.
// ---------------------------------------------------------------------------
__device__ __forceinline__ void compute_chunk(const float (*As)[LDS_PAD],
                                              const float (*Ws)[LDS_PAD],
                                              int wv, int lm, int klo, v8f acc[4])
{
    #pragma unroll
    for (int kk = 0; kk < 32; kk += 4) {
        v2f a;
        a.x = As[wv * 16 + lm][kk + klo];
        a.y = As[wv * 16 + lm][kk + klo + 1];
        #pragma unroll
        for (int nt = 0; nt < 4; ++nt) {
            v2f b;
            b.x = Ws[nt * 16 + lm][kk + klo];
            b.y = Ws[nt * 16 + lm][kk + klo + 1];
            acc[nt] = __builtin_amdgcn_wmma_f32_16x16x4_f32(
                false, a, false, b, (short)0, acc[nt], false, false);
        }
    }
}

// ---------------------------------------------------------------------------
// WMMA-f32 GEMM:  C[M,N] = act(A[M,K] @ W[N,K]^T + bias)
// Block = 128 threads = 4 waves; block tile 64x64; K chunked by 32.
// Async variant: double-buffered LDS, GLOBAL_LOAD_ASYNC_TO_LDS_B128 streams
// the next chunk while WMMAs consume the current one (ASYNCcnt pipelining).
// ---------------------------------------------------------------------------
#if HAVE_ASYNC_LDS

__device__ __forceinline__ void stage_async(const float* __restrict__ A,
                                            const float* __restrict__ W,
                                            float (*As)[LDS_PAD], float (*Ws)[LDS_PAD],
                                            int tid, int m0, int n0,
                                            int M, int N, int K, int k0)
{
    #pragma unroll
    for (int i = 0; i < 4; ++i) {
        int idx = i * 128 + tid;
        int r = idx >> 3, c = (idx & 7) * 4;
        int arow = m0 + r; arow = (arow < M) ? arow : (M - 1);
        __builtin_amdgcn_global_load_async_to_lds_b128(
            (v4i*)&A[(size_t)arow * K + k0 + c],
            (lds_v4i*)&As[r][c], 0, 0);
        int wrow = n0 + r; wrow = (wrow < N) ? wrow : (N - 1);
        __builtin_amdgcn_global_load_async_to_lds_b128(
            (v4i*)&W[(size_t)wrow * K + k0 + c],
            (lds_v4i*)&Ws[r][c], 0, 0);
    }
}

__global__ __launch_bounds__(128)
void k_wmma_gemm(const float* __restrict__ A, const float* __restrict__ W,
                 const float* __restrict__ bias, float* __restrict__ C,
                 int M, int N, int K, int relu)
{
    __shared__ float As[2][64][LDS_PAD];
    __shared__ float Ws[2][64][LDS_PAD];

    const int tid  = threadIdx.x;
    const int lane = tid & 31;
    const int wv   = tid >> 5;
    const int lm   = lane & 15;
    const int klo  = (lane >> 4) * 2;
    const int n0   = blockIdx.x * 64;
    const int m0   = blockIdx.y * 64;
    const int nstages = K >> 5;        // K is a multiple of 32 for all layers

    v8f acc[4] = {};

    stage_async(A, W, As[0], Ws[0], tid, m0, n0, M, N, K, 0);
    if (nstages > 1) {
        stage_async(A, W, As[1], Ws[1], tid, m0, n0, M, N, K, 32);
        WAIT_ASYNC(8);                 // async loads retire in order: stage 0 done
    } else {
        WAIT_ASYNC(0);
    }
    __syncthreads();

    for (int s = 0; s < nstages; ++s) {
        int cur = s & 1;
        compute_chunk(As[cur], Ws[cur], wv, lm, klo, acc);
        if (s + 1 < nstages) {
            __syncthreads();           // all waves done reading buffer 'cur'
            if (s + 2 < nstages) {
                stage_async(A, W, As[cur], Ws[cur], tid, m0, n0, M, N, K, (s + 2) * 32);
                WAIT_ASYNC(8);         // stage s+1 complete (in-order retire)
            } else {
                WAIT_ASYNC(0);
            }
            __syncthreads();           // stage s+1 visible to all waves
        }
    }

    const int rbase = m0 + wv * 16 + (lane >> 4) * 8;
    #pragma unroll
    for (int nt = 0; nt < 4; ++nt) {
        int col = n0 + nt * 16 + lm;
        float bval = (bias != nullptr && col < N) ? bias[col] : 0.0f;
        #pragma unroll
        for (int v = 0; v < 8; ++v) {
            int row = rbase + v;
            if (row < M && col < N) {
                float o = acc[nt][v] + bval;
                if (relu) o = fmaxf(o, 0.0f);
                C[(size_t)row * N + col] = o;
            }
        }
    }
}

#else  // -------- synchronous fallback (proven path) ------------------------

__global__ __launch_bounds__(128)
void k_wmma_gemm(const float* __restrict__ A, const float* __restrict__ W,
                 const float* __restrict__ bias, float* __restrict__ C,
                 int M, int N, int K, int relu)
{
    __shared__ float As[64][LDS_PAD];
    __shared__ float Ws[64][LDS_PAD];

    const int tid  = threadIdx.x;
    const int lane = tid & 31;
    const int wv   = tid >> 5;
    const int lm   = lane & 15;
    const int klo  = (lane >> 4) * 2;
    const int n0   = blockIdx.x * 64;
    const int m0   = blockIdx.y * 64;

    v8f acc[4] = {};

    for (int k0 = 0; k0 < K; k0 += 32) {
        #pragma unroll
        for (int i = 0; i < 4; ++i) {
            int idx = i * 128 + tid;
            int r = idx >> 3, c = (idx & 7) * 4;
            int arow = m0 + r; arow = (arow < M) ? arow : (M - 1);
            *(float4*)&As[r][c] = *(const float4*)&A[(size_t)arow * K + k0 + c];
            int wrow = n0 + r; wrow = (wrow < N) ? wrow : (N - 1);
            *(float4*)&Ws[r][c] = *(const float4*)&W[(size_t)wrow * K + k0 + c];
        }
        if (k0 + 32 < K) {
            int r = tid >> 3, c = (tid & 7) * 4;
            int arow = m0 + r; arow = (arow < M) ? arow : (M - 1);
            __builtin_prefetch(&A[(size_t)arow * K + k0 + 32 + c], 0, 3);
        }
        __syncthreads();
        compute_chunk(As, Ws, wv, lm, klo, acc);
        __syncthreads();
    }

    const int rbase = m0 + wv * 16 + (lane >> 4) * 8;
    #pragma unroll
    for (int nt = 0; nt < 4; ++nt) {
        int col = n0 + nt * 16 + lm;
        float bval = (bias != nullptr && col < N) ? bias[col] : 0.0f;
        #pragma unroll
        for (int v = 0; v < 8; ++v) {
            int row = rbase + v;
            if (row < M && col < N) {
                float o = acc[nt][v] + bval;
                if (relu) o = fmaxf(o, 0.0f);
                C[(size_t)row * N + col] = o;
            }
        }
    }
}

#endif

// ---------------------------------------------------------------------------
// Graph-side kernels
// ---------------------------------------------------------------------------
__global__ void k_deg_init(float* __restrict__ deg, int n)
{
    int i = blockIdx.x * blockDim.x + threadIdx.x;
    if (i < n) deg[i] = 1.0f;   // self-loop contributes +1
}

__global__ void k_deg_edges(float* __restrict__ deg, const int* __restrict__ dst, int e)
{
    int i = blockIdx.x * blockDim.x + threadIdx.x;
    if (i < e) atomicAdd(&deg[dst[i]], 1.0f);
}

__global__ void k_dinv(float* __restrict__ deg, int n)
{
    int i = blockIdx.x * blockDim.x + threadIdx.x;
    if (i < n) deg[i] = rsqrtf(deg[i]);   // in-place deg -> dinv
}

// agg init with self-loop term: agg = hlin * dinv^2 (float4; writes all elems)
__global__ void k_self(float4* __restrict__ agg, const float4* __restrict__ hlin,
                       const float* __restrict__ dinv, int total4)
{
    int i = blockIdx.x * blockDim.x + threadIdx.x;
    if (i >= total4) return;
    float di = dinv[i >> 6];              // HID/4 == 64 float4 per node
    float s = di * di;
    float4 h = hlin[i];
    agg[i] = make_float4(h.x * s, h.y * s, h.z * s, h.w * s);
}

// 4 edges per 256-thread block; 64 threads/edge, float4 gather + 4 L2 atomics
__global__ __launch_bounds__(256)
void k_edge(float* __restrict__ agg, const float* __restrict__ hlin,
            const float* __restrict__ dinv, const int* __restrict__ src,
            const int* __restrict__ dst, int E)
{
    int e = blockIdx.x * 4 + (threadIdx.x >> 6);
    if (e >= E) return;
    int c = (threadIdx.x & 63) * 4;
    int s = src[e], d = dst[e];
    float norm = dinv[s] * dinv[d];
    float4 hv = *(const float4*)&hlin[(size_t)s * HID + c];
    float* base = &agg[(size_t)d * HID + c];
    atomicAdd(base + 0, hv.x * norm);
    atomicAdd(base + 1, hv.y * norm);
    atomicAdd(base + 2, hv.z * norm);
    atomicAdd(base + 3, hv.w * norm);
}

// fused: gcn bias + BN(eval) + relu + optional residual; h updated in place
__global__ void k_bn_relu_res(float4* __restrict__ h, const float4* __restrict__ agg,
                              const float4* __restrict__ gb,
                              const float4* __restrict__ gamma,
                              const float4* __restrict__ beta,
                              const float4* __restrict__ mean,
                              const float4* __restrict__ var,
                              int add_res, int total4)
{
    int i = blockIdx.x * blockDim.x + threadIdx.x;
    if (i >= total4) return;
    int c = i & 63;                        // float4 channel group
    float4 v = agg[i], b = gb[c], g = gamma[c], bt = beta[c];
    float4 mn = mean[c], vr = var[c], r = h[i];
    float ox = fmaxf((v.x + b.x - mn.x) * rsqrtf(vr.x + BN_EPS) * g.x + bt.x, 0.0f);
    float oy = fmaxf((v.y + b.y - mn.y) * rsqrtf(vr.y + BN_EPS) * g.y + bt.y, 0.0f);
    float oz = fmaxf((v.z + b.z - mn.z) * rsqrtf(vr.z + BN_EPS) * g.z + bt.z, 0.0f);
    float ow = fmaxf((v.w + b.w - mn.w) * rsqrtf(vr.w + BN_EPS) * g.w + bt.w, 0.0f);
    if (add_res) { ox += r.x; oy += r.y; oz += r.z; ow += r.w; }
    h[i] = make_float4(ox, oy, oz, ow);
}

// ---------------------------------------------------------------------------
extern "C" void kernel_launch(void* const* d_in, const int* in_sizes, int n_in,
                              void* d_out, int out_size, void* d_ws, size_t ws_size,
                              hipStream_t stream)
{
    (void)in_sizes; (void)n_in; (void)out_size; (void)ws_size;

    const float* x      = (const float*)d_in[0];   // (N, 2048)
    const int*   ei     = (const int*)  d_in[1];   // (2, E)
    const float* in_W   = (const float*)d_in[2];   // (256, 2048)
    const float* in_b   = (const float*)d_in[3];   // (256)
    const float* gcn_W  = (const float*)d_in[4];   // (3, 256, 256)
    const float* gcn_b  = (const float*)d_in[5];   // (3, 256)
    const float* gamma  = (const float*)d_in[6];   // (3, 256)
    const float* beta   = (const float*)d_in[7];
    const float* mean   = (const float*)d_in[8];
    const float* var    = (const float*)d_in[9];
    const float* c1_W   = (const float*)d_in[10];  // (128, 256)
    const float* c1_b   = (const float*)d_in[11];
    const float* c2_W   = (const float*)d_in[12];  // (64, 128)
    const float* c2_b   = (const float*)d_in[13];
    const float* c3_W   = (const float*)d_in[14];  // (10, 64)
    const float* c3_b   = (const float*)d_in[15];
    float* out = (float*)d_out;                    // (N, 10)

    const int* src = ei;
    const int* dst = ei + N_EDGES;

    // workspace carve-up (floats); everything written before read
    float* dinv = (float*)d_ws;                          // 16384 slots (10000 used)
    float* hA   = dinv + 16384;                          // (N, 256) activations / residual
    float* hLin = hA   + (size_t)N_NODES * HID;          // (N, 256) linear output
    float* agg  = hLin + (size_t)N_NODES * HID;          // (N, 256) aggregation
    float* t1   = agg  + (size_t)N_NODES * HID;          // (N, 128)
    float* t2   = t1   + (size_t)N_NODES * 128;          // (N, 64)

    const int total4 = N_NODES * (HID / 4);
    dim3 blk128(128), blk256(256);

    // --- degrees & normalization ---
    k_deg_init <<<(N_NODES + 255) / 256, blk256, 0, stream>>>(dinv, N_NODES);
    k_deg_edges<<<(N_EDGES + 255) / 256, blk256, 0, stream>>>(dinv, dst, N_EDGES);
    k_dinv     <<<(N_NODES + 255) / 256, blk256, 0, stream>>>(dinv, N_NODES);

    // --- input projection + relu : hA = relu(x @ in_W^T + in_b) ---
    {
        dim3 grid((HID + 63) / 64, (N_NODES + 63) / 64);
        k_wmma_gemm<<<grid, blk128, 0, stream>>>(x, in_W, in_b, hA,
                                                 N_NODES, HID, NODE_FEAT, 1);
    }

    // --- GCN layers ---
    for (int L = 0; L < LAYERS; ++L) {
        dim3 grid((HID + 63) / 64, (N_NODES + 63) / 64);
        k_wmma_gemm<<<grid, blk128, 0, stream>>>(hA, gcn_W + (size_t)L * HID * HID,
                                                 nullptr, hLin,
                                                 N_NODES, HID, HID, 0);
        k_self<<<(total4 + 255) / 256, blk256, 0, stream>>>(
            (float4*)agg, (const float4*)hLin, dinv, total4);
        k_edge<<<(N_EDGES + 3) / 4, blk256, 0, stream>>>(agg, hLin, dinv, src, dst, N_EDGES);
        k_bn_relu_res<<<(total4 + 255) / 256, blk256, 0, stream>>>(
            (float4*)hA, (const float4*)agg,
            (const float4*)(gcn_b + (size_t)L * HID),
            (const float4*)(gamma + (size_t)L * HID),
            (const float4*)(beta  + (size_t)L * HID),
            (const float4*)(mean  + (size_t)L * HID),
            (const float4*)(var   + (size_t)L * HID),
            (L > 0) ? 1 : 0, total4);
    }

    // --- classifier head ---
    {
        dim3 g1((128 + 63) / 64, (N_NODES + 63) / 64);
        k_wmma_gemm<<<g1, blk128, 0, stream>>>(hA, c1_W, c1_b, t1,
                                               N_NODES, 128, HID, 1);
        dim3 g2((64 + 63) / 64, (N_NODES + 63) / 64);
        k_wmma_gemm<<<g2, blk128, 0, stream>>>(t1, c2_W, c2_b, t2,
                                               N_NODES, 64, 128, 1);
        dim3 g3((NUM_CLASSES + 63) / 64, (N_NODES + 63) / 64);
        k_wmma_gemm<<<g3, blk128, 0, stream>>>(t2, c3_W, c3_b, out,
                                               N_NODES, NUM_CLASSES, 10 == NUM_CLASSES ? 64 : 64, 0);
    }
}